// katz_pool_60902636257965
// MI455X (gfx1250) — compile-verified
//
#include <hip/hip_runtime.h>
#include <math.h>

#define NN     100000
#define EE     3200000
#define DD     512
#define KSTEPS 3
#define ALPHA  0.1f
#define KTOP   50000

#define TDM_TILE_K   64                 // chunk width (floats)
#define TDM_ROW_PAD  66                 // 64 + 2 pad dwords (TDM pad feature)
#define WAVES_PER_BLK 4

typedef __attribute__((ext_vector_type(2))) float v2f;
typedef __attribute__((ext_vector_type(8))) float v8f;
typedef __attribute__((ext_vector_type(4))) unsigned tdm_g0_t;
typedef __attribute__((ext_vector_type(8))) int      tdm_g1_t;
typedef __attribute__((ext_vector_type(4))) int      tdm_g2_t;

__device__ __forceinline__ unsigned rfl_u32(unsigned v) {
  return (unsigned)__builtin_amdgcn_readfirstlane((int)v);
}

// Issue one TDM load: 16 rows x 64 cols fp32 tile of x into LDS, with
// 2-dword padding after every 64 dwords (row stride 66 floats in LDS).
__device__ __forceinline__ void tdm_load_tile(unsigned lds_off,
                                              unsigned long long gaddr) {
  tdm_g0_t g0;
  g0[0] = 1u;                                   // count=1, user descriptor
  g0[1] = lds_off;                              // lds_addr (bytes)
  g0[2] = (unsigned)gaddr;                      // global_addr[31:0]
  g0[3] = ((unsigned)(gaddr >> 32) & 0x01FFFFFFu) | (2u << 30);  // type=2
  tdm_g1_t g1;
  // data_size=2(4B) | pad_enable | pad_interval=5(64dw) | pad_amount=1(2dw)
  g1[0] = (int)0x03520000u;
  g1[1] = (int)(512u << 16);                    // tensor_dim0 = 512
  g1[2] = (int)((100000u & 0xFFFFu) << 16);     // tensor_dim1 lo = N
  g1[3] = (int)((64u << 16) | (100000u >> 16)); // tile_dim0=64 | tensor_dim1 hi
  g1[4] = 16;                                   // tile_dim1 = 16 rows
  g1[5] = 512;                                  // tensor_dim0_stride = 512
  g1[6] = 0;
  g1[7] = 0;
  tdm_g2_t g2 = {};                             // 2-D tensor: groups 2/3 unused
  tdm_g2_t g3 = {};
  tdm_g1_t g4 = {};                             // clang-23 6-arg form: extra group
  __builtin_amdgcn_tensor_load_to_lds(g0, g1, g2, g3, g4, 0);
}

// ---------------------------------------------------------------------------
// Stage 1: s0 = x @ W + b  via V_WMMA_F32_16X16X4_F32, A staged through LDS by
// the Tensor Data Mover (double-buffered, s_wait_tensorcnt synchronized).
// One wave computes 16 rows. A-matrix layout (ISA 7.12.2, 32-bit 16x4): lanes
// 0-15 hold K=0,1 in v0,v1; lanes 16-31 hold K=2,3. B (4x16) is W broadcast to
// all 16 columns, so C vgpr r at lane 0 = row r, at lane 16 = row 8+r.
// ---------------------------------------------------------------------------
__global__ __launch_bounds__(WAVES_PER_BLK * 32) void score_wmma_kernel(
    const float* __restrict__ x, const float* __restrict__ W,
    const float* __restrict__ bias, float* __restrict__ score,
    float* __restrict__ cur) {
  __shared__ float sW[DD];
  __shared__ float sx[WAVES_PER_BLK][2][16 * TDM_ROW_PAD];

  // stage W cooperatively BEFORE any wave can exit (single safe barrier)
  for (int i = threadIdx.x; i < DD; i += blockDim.x) sW[i] = W[i];
  __syncthreads();

  int wid  = threadIdx.x >> 5;
  int lane = threadIdx.x & 31;
  int base = (blockIdx.x * WAVES_PER_BLK + wid) * 16;  // 16 rows per wave
  if (base >= NN) return;                              // wave-uniform exit

  int khalf = (lane >> 4) << 1;                        // 0 | 2
  const float* arow_base = &sx[wid][0][0] + (lane & 15) * TDM_ROW_PAD + khalf;
  unsigned lbuf0 = rfl_u32((unsigned)(size_t)(void*)&sx[wid][0][0]);
  unsigned lbuf1 = rfl_u32((unsigned)(size_t)(void*)&sx[wid][1][0]);
  unsigned long long gbase =
      (unsigned long long)(size_t)(const void*)(x + (size_t)base * DD);
  unsigned g_lo = rfl_u32((unsigned)gbase);
  unsigned g_hi = rfl_u32((unsigned)(gbase >> 32));
  unsigned long long g = ((unsigned long long)g_hi << 32) | g_lo;

  const int CHUNKS = DD / TDM_TILE_K;  // 8
  tdm_load_tile(lbuf0, g);             // prime buffer 0

  v8f c = {};
  for (int ch = 0; ch < CHUNKS; ++ch) {
    if (ch + 1 < CHUNKS) {
      tdm_load_tile((ch & 1) ? lbuf0 : lbuf1,
                    g + (unsigned long long)(ch + 1) * TDM_TILE_K * 4ull);
      __builtin_amdgcn_s_wait_tensorcnt(1);  // current chunk's tile is in LDS
    } else {
      __builtin_amdgcn_s_wait_tensorcnt(0);
    }
    const float* arow = arow_base + (ch & 1) * (16 * TDM_ROW_PAD);
    const float* wch  = sW + ch * TDM_TILE_K + khalf;
#pragma unroll
    for (int kk = 0; kk < TDM_TILE_K; kk += 4) {
      v2f a = *(const v2f*)(arow + kk);   // ds_load_b64, pad kills conflicts
      v2f bb;
      bb.x = wch[kk];
      bb.y = wch[kk + 1];
      c = __builtin_amdgcn_wmma_f32_16x16x4_f32(
          /*neg_a=*/false, a, /*neg_b=*/false, bb,
          /*c_mod=*/(short)0, c, /*reuse_a=*/false, /*reuse_b=*/false);
    }
  }

  float bv = bias[0];
  if (lane == 0 || lane == 16) {
    int mbase = base + ((lane >> 4) << 3);
#pragma unroll
    for (int r = 0; r < 8; ++r) {
      float s = c[r] + bv;
      score[mbase + r] = s;
      cur[mbase + r]   = s;
    }
  }
}

// ---------------------------------------------------------------------------
// Stage 2: katz propagation. nxt[dst] += cur[src]; then
// cur = alpha*nxt; score += cur; nxt = 0.
// ---------------------------------------------------------------------------
__global__ void edge_prop_kernel(const long long* __restrict__ src,
                                 const long long* __restrict__ dst,
                                 const float* __restrict__ cur,
                                 float* __restrict__ nxt, int ne) {
  int e = blockIdx.x * blockDim.x + threadIdx.x;
  if (e < ne) {
    int s = (int)src[e];
    int d = (int)dst[e];
    atomicAdd(&nxt[d], cur[s]);
  }
}

__global__ void node_update_kernel(float* __restrict__ score,
                                   float* __restrict__ cur,
                                   float* __restrict__ nxt, int n) {
  int i = blockIdx.x * blockDim.x + threadIdx.x;
  if (i < n) {
    float v = ALPHA * nxt[i];
    cur[i] = v;
    score[i] += v;
    nxt[i] = 0.0f;
  }
}

__global__ void zero_nxt_kernel(float* __restrict__ nxt, int n) {
  int i = blockIdx.x * blockDim.x + threadIdx.x;
  if (i < n) nxt[i] = 0.0f;
}

// ---------------------------------------------------------------------------
// Stage 3: exact top-K via 4-round MSB radix select on monotone uint keys.
// ---------------------------------------------------------------------------
__global__ void make_keys_kernel(const float* __restrict__ score,
                                 unsigned* __restrict__ keys, int n) {
  int i = blockIdx.x * blockDim.x + threadIdx.x;
  if (i < n) {
    unsigned u = __float_as_uint(score[i]);
    u = (u & 0x80000000u) ? ~u : (u | 0x80000000u);
    keys[i] = u;
  }
}

__global__ void init_state_kernel(unsigned* state) {
  state[0] = 0u;
  state[1] = (unsigned)KTOP;
}

__global__ void zero_hist_kernel(unsigned* hist) { hist[threadIdx.x] = 0u; }

__global__ void hist_kernel(const unsigned* __restrict__ keys,
                            unsigned* __restrict__ hist,
                            const unsigned* __restrict__ state,
                            int shift, int n) {
  __shared__ unsigned lh[256];
  if (threadIdx.x < 256) lh[threadIdx.x] = 0u;
  __syncthreads();
  int i = blockIdx.x * blockDim.x + threadIdx.x;
  if (i < n) {
    unsigned k = keys[i];
    unsigned prefix = state[0];
    bool match = (shift == 24) || ((k >> (shift + 8)) == (prefix >> (shift + 8)));
    if (match) atomicAdd(&lh[(k >> shift) & 255u], 1u);
  }
  __syncthreads();
  if (threadIdx.x < 256 && lh[threadIdx.x]) atomicAdd(&hist[threadIdx.x], lh[threadIdx.x]);
}

__global__ void select_kernel(const unsigned* __restrict__ hist,
                              unsigned* __restrict__ state, int shift) {
  if (threadIdx.x == 0) {
    unsigned R = state[1];
    unsigned cum = 0;
    int b = 255;
    for (; b >= 0; --b) {
      unsigned c = hist[b];
      if (cum + c >= R) break;
      cum += c;
    }
    if (b < 0) b = 0;
    state[0] |= ((unsigned)b) << shift;
    state[1] = R - cum;
  }
}

// ---------------------------------------------------------------------------
// Deterministic compaction: block counts -> serial scan -> ordered scatter.
// ---------------------------------------------------------------------------
__global__ void count_kernel(const unsigned* __restrict__ keys,
                             const unsigned* __restrict__ state,
                             unsigned* __restrict__ blkA,
                             unsigned* __restrict__ blkB,
                             int* __restrict__ new_id, int n) {
  __shared__ unsigned wA[8], wB[8];
  int i    = blockIdx.x * blockDim.x + threadIdx.x;
  int lane = threadIdx.x & 31;
  int wid  = threadIdx.x >> 5;
  unsigned T = state[0];
  bool above = false, equal = false;
  if (i < n) {
    unsigned k = keys[i];
    above = k > T;
    equal = (k == T);
    new_id[i] = -1;
  }
  unsigned long long mA = __ballot(above);
  unsigned long long mB = __ballot(equal);
  if (lane == 0) { wA[wid] = (unsigned)__popcll(mA); wB[wid] = (unsigned)__popcll(mB); }
  __syncthreads();
  if (threadIdx.x == 0) {
    unsigned a = 0, b2 = 0;
    for (int w = 0; w < 8; ++w) { a += wA[w]; b2 += wB[w]; }
    blkA[blockIdx.x] = a;
    blkB[blockIdx.x] = b2;
  }
}

__global__ void scan_blocks_kernel(unsigned* blkA, unsigned* blkB, int nb) {
  if (threadIdx.x == 0 && blockIdx.x == 0) {
    unsigned run = 0;
    for (int b = 0; b < nb; ++b) { unsigned t = blkA[b]; blkA[b] = run; run += t; }
    unsigned CA = run;
    unsigned run2 = 0;
    for (int b = 0; b < nb; ++b) { unsigned t = blkB[b]; blkB[b] = CA + run2; run2 += t; }
  }
}

__global__ void scatter_kernel(const unsigned* __restrict__ keys,
                               const unsigned* __restrict__ state,
                               const unsigned* __restrict__ offA,
                               const unsigned* __restrict__ offB,
                               int* __restrict__ new_id,
                               int* __restrict__ perm, int n) {
  __shared__ unsigned wA[8], wB[8], pA[8], pB[8];
  int i    = blockIdx.x * blockDim.x + threadIdx.x;
  int lane = threadIdx.x & 31;
  int wid  = threadIdx.x >> 5;
  unsigned T = state[0];
  bool above = false, equal = false;
  if (i < n) {
    unsigned k = keys[i];
    above = k > T;
    equal = (k == T);
  }
  unsigned long long mA = __ballot(above);
  unsigned long long mB = __ballot(equal);
  if (lane == 0) { wA[wid] = (unsigned)__popcll(mA); wB[wid] = (unsigned)__popcll(mB); }
  __syncthreads();
  if (threadIdx.x == 0) {
    unsigned ra = 0, rb = 0;
    for (int w = 0; w < 8; ++w) { pA[w] = ra; ra += wA[w]; pB[w] = rb; rb += wB[w]; }
  }
  __syncthreads();
  if (i < n) {
    unsigned long long below = (1ull << lane) - 1ull;
    if (above) {
      int pos = (int)(offA[blockIdx.x] + pA[wid] + (unsigned)__popcll(mA & below));
      new_id[i] = pos;
      perm[pos] = i;
    } else if (equal) {
      int pos = (int)(offB[blockIdx.x] + pB[wid] + (unsigned)__popcll(mB & below));
      if (pos < KTOP) {
        new_id[i] = pos;
        perm[pos] = i;
      }
    }
  }
}

// ---------------------------------------------------------------------------
// Stage 4: x_out[r] = x[perm[r]] * tanh(score[perm[r]])   (B128 streams)
// ---------------------------------------------------------------------------
__global__ void gather_kernel(const float* __restrict__ x,
                              const float* __restrict__ score,
                              const int* __restrict__ perm,
                              float* __restrict__ xout) {
  long long t = (long long)blockIdx.x * blockDim.x + threadIdx.x;
  if (t >= (long long)KTOP * (DD / 4)) return;
  int r  = (int)(t >> 7);
  int c  = (int)(t & 127);
  int sr = perm[r];
  float s = tanhf(score[sr]);
  const float4* xin = (const float4*)(x + (size_t)sr * DD);
  __builtin_prefetch(xin + ((c + 32) & 127), 0, 0);  // global_prefetch_b8
  float4 v = xin[c];
  float4 o;
  o.x = v.x * s; o.y = v.y * s; o.z = v.z * s; o.w = v.w * s;
  ((float4*)(xout + (size_t)r * DD))[c] = o;
}

// ---------------------------------------------------------------------------
// Stage 5: edge remap + validity mask.
// ---------------------------------------------------------------------------
__global__ void remap_kernel(const long long* __restrict__ src,
                             const long long* __restrict__ dst,
                             const int* __restrict__ new_id,
                             long long* __restrict__ out_edges,
                             unsigned char* __restrict__ out_mask, int ne) {
  int e = blockIdx.x * blockDim.x + threadIdx.x;
  if (e < ne) {
    long long r = new_id[(int)src[e]];
    long long c = new_id[(int)dst[e]];
    bool m = (r >= 0) && (c >= 0);
    out_edges[e]      = m ? r : -1ll;
    out_edges[ne + e] = m ? c : -1ll;
    out_mask[e] = (unsigned char)m;
  }
}

// ---------------------------------------------------------------------------
extern "C" void kernel_launch(void* const* d_in, const int* in_sizes, int n_in,
                              void* d_out, int out_size, void* d_ws, size_t ws_size,
                              hipStream_t stream) {
  const float*     x  = (const float*)d_in[0];
  const long long* ei = (const long long*)d_in[1];  // int64 [2, E]
  const float*     W  = (const float*)d_in[2];
  const float*     b  = (const float*)d_in[3];
  const long long* src = ei;
  const long long* dst = ei + EE;

  // workspace layout (bytes)
  char* ws = (char*)d_ws;
  float*    score  = (float*)(ws + 0);
  float*    cur    = (float*)(ws + 400000);
  float*    nxt    = (float*)(ws + 800000);
  unsigned* keys   = (unsigned*)(ws + 1200000);
  int*      new_id = (int*)(ws + 1600000);
  unsigned* hist   = (unsigned*)(ws + 2000000);
  unsigned* state  = (unsigned*)(ws + 2001024);
  unsigned* blkA   = (unsigned*)(ws + 2001152);
  unsigned* blkB   = (unsigned*)(ws + 2002752);

  // output layout: x_out f32 | new_edge_index i64 | edge_mask u8 | perm i32
  char* out = (char*)d_out;
  float*         x_out     = (float*)out;
  long long*     out_edges = (long long*)(out + 102400000);
  unsigned char* out_mask  = (unsigned char*)(out + 153600000);
  int*           perm      = (int*)(out + 156800000);

  dim3 blk(256);
  const int nbN = (NN + 255) / 256;  // 391
  const int nbE = (EE + 255) / 256;  // 12500

  zero_nxt_kernel<<<nbN, blk, 0, stream>>>(nxt, NN);
  init_state_kernel<<<1, 1, 0, stream>>>(state);

  // GEMV scores via TDM-staged fp32 WMMA: one wave = 16 rows, 4 waves/block.
  int nWaves = NN / 16;  // 6250
  int nBlksScore = (nWaves + WAVES_PER_BLK - 1) / WAVES_PER_BLK;  // 1563
  score_wmma_kernel<<<nBlksScore, WAVES_PER_BLK * 32, 0, stream>>>(x, W, b, score, cur);

  for (int k = 0; k < KSTEPS; ++k) {
    edge_prop_kernel<<<nbE, blk, 0, stream>>>(src, dst, cur, nxt, EE);
    node_update_kernel<<<nbN, blk, 0, stream>>>(score, cur, nxt, NN);
  }

  make_keys_kernel<<<nbN, blk, 0, stream>>>(score, keys, NN);
  for (int r = 0; r < 4; ++r) {
    int shift = 24 - 8 * r;
    zero_hist_kernel<<<1, 256, 0, stream>>>(hist);
    hist_kernel<<<nbN, blk, 0, stream>>>(keys, hist, state, shift, NN);
    select_kernel<<<1, 1, 0, stream>>>(hist, state, shift);
  }

  count_kernel<<<nbN, blk, 0, stream>>>(keys, state, blkA, blkB, new_id, NN);
  scan_blocks_kernel<<<1, 1, 0, stream>>>(blkA, blkB, nbN);
  scatter_kernel<<<nbN, blk, 0, stream>>>(keys, state, blkA, blkB, new_id, perm, NN);

  long long gthreads = (long long)KTOP * (DD / 4);
  gather_kernel<<<(int)((gthreads + 255) / 256), blk, 0, stream>>>(x, score, perm, x_out);
  remap_kernel<<<nbE, blk, 0, stream>>>(src, dst, new_id, out_edges, out_mask, EE);
}